// RoiPooling_65601330479347
// MI455X (gfx1250) — compile-verified
//
#include <hip/hip_runtime.h>
#include <cstdint>

// ---------------------------------------------------------------------------
// ROI bilinear pooling, MI455X (gfx1250).
//
// Memory-bound op: 103 MB of output stores (HBM floor ~4.4us @ 23.3 TB/s),
// 16 MB feature map fully L2-resident. WMMA is intentionally NOT used: the
// bin-blend matrix has 4 nonzeros/row, so dense f32 WMMA would add ~10 GFLOP
// of wasted work through the slow 16x16x4_f32 shape and blow past the memory
// floor. Instead we use the CDNA5 async global->LDS DMA path to pipeline the
// 4-corner-row gathers (per-wave private double buffer, no barriers), and
// non-temporal b128 stores so the output stream doesn't evict the L2-resident
// feature map.
// ---------------------------------------------------------------------------

typedef float v4f __attribute__((ext_vector_type(4)));

#define POOL    7
#define HFEAT   64
#define WFEAT   64
#define CFEAT   1024
#define THREADS 256            // 8 wave32s; each thread owns a float4 slice

__device__ __forceinline__ void axis_coords(int lo, int hi, int t,
                                            int& c0, int& c1, float& frac) {
    // Matches reference: src = t * (in_size / POOL); i0 = floor(src);
    // i1 = min(i0+1, hi-lo); coords offset by lo.
    const float in_size = (float)(hi - lo + 1);
    const float src     = (float)t * (in_size / (float)POOL);
    const int   i0      = (int)floorf(src);
    frac = src - (float)i0;
    int i1 = i0 + 1;
    const int cap = hi - lo;
    if (i1 > cap) i1 = cap;
    c0 = lo + i0;
    c1 = lo + i1;
}

__global__ __launch_bounds__(THREADS)
void roi_pool_async_kernel(const float* __restrict__ feat,
                           const float* __restrict__ rois,
                           const int*   __restrict__ stride_ptr,
                           float*       __restrict__ out,
                           int nrois) {
    // Double-buffered staging: 2 buffers x 4 corner rows x 1024 f32 = 32 KB.
    __shared__ __align__(16) float smem[2 * 4 * CFEAT];

    const int n = blockIdx.x / POOL;   // ROI index
    const int i = blockIdx.x % POOL;   // pooled row
    if (n >= nrois) return;            // never taken (exact grid), keeps EXEC=all-1s

    const float stride = (float)stride_ptr[0];
    // ROI box -> integer feature coords (truncation, matches astype(int32) on +ve)
    const float r0 = rois[n * 4 + 0];
    const float r1 = rois[n * 4 + 1];
    const float r2 = rois[n * 4 + 2];
    const float r3 = rois[n * 4 + 3];
    const int ymin = (int)(r0 / stride);
    const int xmin = (int)(r1 / stride);
    const int ymax = (int)(r2 / stride);
    const int xmax = (int)(r3 / stride);

    int y0, y1; float dy;
    axis_coords(ymin, ymax, i, y0, y1, dy);

    int   x0[POOL], x1[POOL];
    float dx[POOL];
#pragma unroll
    for (int j = 0; j < POOL; ++j) axis_coords(xmin, xmax, j, x0[j], x1[j], dx[j]);

    const int      tid      = threadIdx.x;
    const unsigned cbyte    = (unsigned)tid * 16u;                 // this lane's 16B slice
    // Low 32 bits of the generic LDS address == byte offset inside the
    // workgroup's LDS allocation (shared aperture is 4GB-aligned).
    const unsigned lds_base = (unsigned)(uintptr_t)(&smem[0]);

    const char* fbase = (const char*)feat;

    // Issue 4 async b128 DMA loads (one per corner row) for bin j into buf[j&1].
    // Each lane stages exactly the 16B it will read back -> per-wave private
    // pipeline, no workgroup barrier required.
    auto stage = [&](int j) {
        const unsigned buf = lds_base + (unsigned)(j & 1) * (4u * CFEAT * 4u);
#pragma unroll
        for (int r = 0; r < 4; ++r) {
            const int yy = (r < 2) ? y0 : y1;
            const int xx = (r & 1) ? x1[j] : x0[j];
            const char* g = fbase
                          + ((size_t)(yy * WFEAT + xx) * CFEAT) * sizeof(float)
                          + cbyte;
            const unsigned l = buf + (unsigned)r * (CFEAT * 4u) + cbyte;
            asm volatile("global_load_async_to_lds_b128 %0, %1, off"
                         :: "v"(l), "v"(g)
                         : "memory");
        }
    };

    stage(0);

    float* outp = out + (size_t)n * (POOL * POOL * CFEAT)
                      + (size_t)i * (POOL * CFEAT);

#pragma unroll
    for (int j = 0; j < POOL; ++j) {
        if (j + 1 < POOL) {
            // WAR fence: our previous-iteration LDS reads must have completed
            // before the DMA engine may overwrite that buffer two stages on.
            asm volatile("s_wait_dscnt 0x0" ::: "memory");
            stage(j + 1);                                   // 8 asyncs in flight
            // Per-wave asyncs retire in order: cnt<=4 => stage(j)'s 4 landed,
            // while stage(j+1)'s 4 keep flying (latency hiding).
            asm volatile("s_wait_asynccnt 0x4" ::: "memory");
        } else {
            asm volatile("s_wait_asynccnt 0x0" ::: "memory");
        }

        const float* buf = &smem[(j & 1) * 4 * CFEAT];
        const v4f f00 = *(const v4f*)(buf + 0 * CFEAT + tid * 4);
        const v4f f01 = *(const v4f*)(buf + 1 * CFEAT + tid * 4);
        const v4f f10 = *(const v4f*)(buf + 2 * CFEAT + tid * 4);
        const v4f f11 = *(const v4f*)(buf + 3 * CFEAT + tid * 4);

        const v4f top = f00 + (f01 - f00) * dx[j];
        const v4f bot = f10 + (f11 - f10) * dx[j];
        const v4f res = top + (bot - top) * dy;

        // NT store: 103MB one-shot output stream must not evict the
        // L2-resident 16MB feature map.
        __builtin_nontemporal_store(res, (v4f*)(outp + (size_t)j * CFEAT + tid * 4));
    }
}

extern "C" void kernel_launch(void* const* d_in, const int* in_sizes, int n_in,
                              void* d_out, int out_size, void* d_ws, size_t ws_size,
                              hipStream_t stream) {
    const float* feat       = (const float*)d_in[0];  // (1,64,64,1024) f32, NHWC
    const float* rois       = (const float*)d_in[1];  // (N,4) f32
    const int*   stride_ptr = (const int*)d_in[2];    // scalar int (16)
    float*       out        = (float*)d_out;          // (N, 49*1024) f32

    const int nrois = in_sizes[1] / 4;                // 512
    dim3 grid((unsigned)(nrois * POOL));              // one block per (roi, i-row)
    roi_pool_async_kernel<<<grid, THREADS, 0, stream>>>(feat, rois, stride_ptr,
                                                        out, nrois);
}